// MCLET_28037546509013
// MI455X (gfx1250) — compile-verified
//
#include <hip/hip_runtime.h>
#include <math.h>

// ---------------- problem constants (match reference) ----------------
#define NE_     100000
#define NT_     1024
#define NC_     128
#define D_      128
#define NR_     500
#define H_      8
#define DH_     128
#define E_      4
#define B_      512
#define DEG_    8
#define NSRC_   4096        // B*DEG
#define T_      1024        // NT
#define NNODES_ (NE_ + NT_ + NC_)
#define INV_TAU 2.0f        // 1/0.5
#define DECAY_  1e-4f
#define CLW_    0.1f

typedef __attribute__((ext_vector_type(2))) float fx2;
typedef __attribute__((ext_vector_type(8))) float fx8;

// wait for all outstanding async global<->LDS transfers of this wave
#define WAIT_ASYNC() asm volatile("s_wait_asynccnt 0" ::: "memory")

// block-relative LDS byte offset from a generic pointer to a __shared__ object
// (flat LDS address = {shared_aperture, 32'offset}; low 32 bits are the offset)
__device__ __forceinline__ unsigned lds_off(const void* p) {
    return (unsigned)(size_t)p;
}

// =====================================================================
// LightGCN: init / scatter / accumulate / finalize(+LayerNorm)
// =====================================================================
__global__ void gcn_init(const float* __restrict__ a, long long nA,
                         const float* __restrict__ b, long long nB,
                         float* __restrict__ x, float* __restrict__ acc) {
    long long i = (long long)blockIdx.x * blockDim.x + threadIdx.x;
    if (i < nA)            { float v = a[i];      x[i] = v; acc[i] = v; }
    else if (i < nA + nB)  { float v = b[i - nA]; x[i] = v; acc[i] = v; }
}

// one wave (32 lanes) per edge; each lane handles 4 feature dims
__global__ void gcn_scatter(const float* __restrict__ x, float* __restrict__ y,
                            const int* __restrict__ row, const int* __restrict__ col,
                            const float* __restrict__ val, int nnz) {
    long long t = (long long)blockIdx.x * blockDim.x + threadIdx.x;
    int e = (int)(t >> 5);
    if (e >= nnz) return;
    int lane = (int)(t & 31);
    float v = val[e];
    const float* xs = x + (size_t)col[e] * D_;
    float*       yd = y + (size_t)row[e] * D_;
    int d = lane * 4;
    float4 xv = *(const float4*)(xs + d);
    atomicAdd(yd + d + 0, v * xv.x);
    atomicAdd(yd + d + 1, v * xv.y);
    atomicAdd(yd + d + 2, v * xv.z);
    atomicAdd(yd + d + 3, v * xv.w);
}

__global__ void vec_add(float* __restrict__ acc, const float* __restrict__ y, long long n) {
    long long i = (long long)blockIdx.x * blockDim.x + threadIdx.x;
    if (i < n) acc[i] += y[i];
}

// m = acc/3 then LayerNorm(row); route rows < na to dstA, rest to dstB
__global__ __launch_bounds__(128) void gcn_finalize_ln(
        const float* __restrict__ acc, int na,
        const float* __restrict__ g, const float* __restrict__ bvec,
        float* __restrict__ dstA, float* __restrict__ dstB) {
    int r = blockIdx.x, t = threadIdx.x;
    float v = acc[(size_t)r * D_ + t] * (1.0f / 3.0f);
    __shared__ float red[128];
    red[t] = v; __syncthreads();
    for (int s = 64; s > 0; s >>= 1) { if (t < s) red[t] += red[t + s]; __syncthreads(); }
    float mu = red[0] * (1.0f / 128.0f);
    __syncthreads();
    float dv = v - mu;
    red[t] = dv * dv; __syncthreads();
    for (int s = 64; s > 0; s >>= 1) { if (t < s) red[t] += red[t + s]; __syncthreads(); }
    float var = red[0] * (1.0f / 128.0f);
    float o = dv * rsqrtf(var + 1e-5f) * g[t] + bvec[t];
    float* dst = (r < na) ? (dstA + (size_t)r * D_) : (dstB + (size_t)(r - na) * D_);
    dst[t] = o;
}

// =====================================================================
// Generic fp32 WMMA GEMM:  C = act(A[M,K] @ B[K,N] + bias)
//   block = 256 thr (8 waves), tile 64(M) x 32(N), K-step 16.
//   Double-buffered LDS staging via GLOBAL_LOAD_ASYNC_TO_LDS (ASYNCcnt):
//   each thread issues one b128 (A) and one b64 (B) async load per K-step;
//   next tile's DMA runs under the current tile's chained WMMAs.
//   act: 0 = none, 1 = elu.  expert >= 0 -> MoE accumulate epilogue:
//        C[row,col] += gate[row*4+expert] * (acc + bias[col])
// Requires M%64==0, N%32==0, K%16==0 (true for all uses here).
// =====================================================================
#define GEMM_BM 64
#define GEMM_BN 32
#define GEMM_BK 16
#define LDA_S   20          // A staged m-major [64][20]; 16B-aligned float4 slots
#define LDB_S   40          // B staged k-major [16][40]; halves in disjoint banks

__global__ __launch_bounds__(256) void gemm_wmma_f32(
        const float* __restrict__ A, int lda,
        const float* __restrict__ Bm, int ldb,
        const float* __restrict__ bias,
        float* __restrict__ C, int ldc,
        int M, int N, int K, int act,
        const float* __restrict__ gate, int expert) {
    __shared__ float As[2][GEMM_BM * LDA_S];
    __shared__ float Bs[2][GEMM_BK * LDB_S];
    int tid  = threadIdx.x;
    int wave = tid >> 5, lane = tid & 31;
    int lnlo = lane & 15, lnhi = lane >> 4;
    int tm = wave >> 1, tn = wave & 1;        // 4x2 wave grid of 16x16 tiles
    int m0 = blockIdx.y * GEMM_BM;
    int n0 = blockIdx.x * GEMM_BN;

    // staging assignment: A tile 64x16 -> thread t: row am, 4 k's; B tile 16x32
    int am  = tid >> 2;
    int akq = (tid & 3) * 4;
    int bkk = tid >> 4;
    int bnn = (tid & 15) * 2;
    unsigned aoff[2] = { lds_off(&As[0][am * LDA_S + akq]),
                         lds_off(&As[1][am * LDA_S + akq]) };
    unsigned boff[2] = { lds_off(&Bs[0][bkk * LDB_S + bnn]),
                         lds_off(&Bs[1][bkk * LDB_S + bnn]) };
    const float* gA0 = A  + (size_t)(m0 + am) * lda + akq;
    const float* gB0 = Bm + (size_t)bkk * ldb + n0 + bnn;

    auto stage = [&](int bi, int kk) {
        const float* gA = gA0 + kk;
        asm volatile("global_load_async_to_lds_b128 %0, %1, off"
                     :: "v"(aoff[bi]), "v"(gA) : "memory");
        const float* gB = gB0 + (size_t)kk * ldb;
        asm volatile("global_load_async_to_lds_b64 %0, %1, off"
                     :: "v"(boff[bi]), "v"(gB) : "memory");
    };

    fx8 acc = {};
    int arow = (tm * 16 + lnlo) * LDA_S;
    int bcol = tn * 16 + lnlo;
    int nk = K / GEMM_BK;
    stage(0, 0);
    for (int i = 0; i < nk; ++i) {
        int cur = i & 1;
        WAIT_ASYNC();
        __syncthreads();                      // tile `cur` resident for all waves
        if (i + 1 < nk) stage(cur ^ 1, (i + 1) * GEMM_BK);
#pragma unroll
        for (int k4 = 0; k4 < GEMM_BK / 4; ++k4) {
            int kb = k4 * 4 + lnhi * 2;
            fx2 af = { As[cur][arow + kb], As[cur][arow + kb + 1] };   // ds_load_b64
            fx2 bf = { Bs[cur][(kb + 0) * LDB_S + bcol],
                       Bs[cur][(kb + 1) * LDB_S + bcol] };
            acc = __builtin_amdgcn_wmma_f32_16x16x4_f32(false, af, false, bf,
                                                        (short)0, acc, false, false);
        }
    }
    int col = n0 + tn * 16 + lnlo;
    float bb = bias ? bias[col] : 0.0f;
#pragma unroll
    for (int r = 0; r < 8; ++r) {
        int row = m0 + tm * 16 + r + 8 * lnhi;
        float v = acc[r] + bb;
        if (act == 1) v = (v > 0.0f) ? v : (expf(v) - 1.0f);  // elu
        if (expert >= 0) {
            C[(size_t)row * ldc + col] += gate[row * 4 + expert] * v;
        } else {
            C[(size_t)row * ldc + col] = v;
        }
    }
}

// =====================================================================
// Contrastive similarity: per 16x16 tile, WMMA s11/s12/s22 over K=128,
// exp(./tau), reduce rows (shfl butterflies) + col sums of s12 + diags.
// =====================================================================
__global__ __launch_bounds__(32) void cl_sim(
        const float* __restrict__ z1, const float* __restrict__ z2,
        float* __restrict__ r11, float* __restrict__ r12, float* __restrict__ r22,
        float* __restrict__ c12, float* __restrict__ d11, float* __restrict__ d12,
        float* __restrict__ d22) {
    int lane = threadIdx.x;
    int lnlo = lane & 15, lnhi = lane >> 4;
    int i0 = blockIdx.y * 16, j0 = blockIdx.x * 16;
    fx8 c11 = {}, cx = {}, c22 = {};
    const float* z1i = z1 + (size_t)(i0 + lnlo) * D_;
    const float* z2i = z2 + (size_t)(i0 + lnlo) * D_;
    const float* z1j = z1 + (size_t)(j0 + lnlo) * D_;
    const float* z2j = z2 + (size_t)(j0 + lnlo) * D_;
    for (int k = 0; k < D_; k += 4) {
        int kb = k + lnhi * 2;
        fx2 a1 = { z1i[kb], z1i[kb + 1] };
        fx2 a2 = { z2i[kb], z2i[kb + 1] };
        fx2 b1 = { z1j[kb], z1j[kb + 1] };   // B[k][n] = z1[j0+n][k]
        fx2 b2 = { z2j[kb], z2j[kb + 1] };
        c11 = __builtin_amdgcn_wmma_f32_16x16x4_f32(false, a1, false, b1, (short)0, c11, false, false);
        cx  = __builtin_amdgcn_wmma_f32_16x16x4_f32(false, a1, false, b2, (short)0, cx,  false, false);
        c22 = __builtin_amdgcn_wmma_f32_16x16x4_f32(false, a2, false, b2, (short)0, c22, false, false);
    }
    float colp = 0.0f;
#pragma unroll
    for (int r = 0; r < 8; ++r) {
        int Mi = r + 8 * lnhi;                 // row within tile for this vgpr/lane
        float e11 = expf(c11[r] * INV_TAU);
        float e12 = expf(cx[r]  * INV_TAU);
        float e22 = expf(c22[r] * INV_TAU);
        float s11 = e11, s12 = e12, s22 = e22; // row sums across the 16 N-lanes
        for (int off = 1; off < 16; off <<= 1) {
            s11 += __shfl_xor(s11, off, 32);
            s12 += __shfl_xor(s12, off, 32);
            s22 += __shfl_xor(s22, off, 32);
        }
        if (lnlo == 0) {
            atomicAdd(&r11[i0 + Mi], s11);
            atomicAdd(&r12[i0 + Mi], s12);
            atomicAdd(&r22[i0 + Mi], s22);
        }
        colp += e12;                           // col-sum partial (fixed N=lnlo)
        if (i0 == j0 && lnlo == Mi) {          // tile diagonal
            d11[i0 + Mi] = e11;
            d12[i0 + Mi] = e12;
            d22[i0 + Mi] = e22;
        }
    }
    float cother = __shfl_xor(colp, 16, 32);
    if (lnhi == 0) atomicAdd(&c12[j0 + lnlo], colp + cother);
}

__global__ __launch_bounds__(256) void cl_finalize(
        const float* __restrict__ r11, const float* __restrict__ d11,
        const float* __restrict__ r12, const float* __restrict__ r22,
        const float* __restrict__ d22, const float* __restrict__ c12,
        const float* __restrict__ d12, float* __restrict__ accum) {
    int i = blockIdx.x * 256 + threadIdx.x;
    float l = 0.0f;
    if (i < NSRC_) {
        float den1 = r11[i] - d11[i] + r12[i];
        float den2 = r22[i] - d22[i] + c12[i];
        float dd = d12[i];
        l = 0.5f * (-logf(dd / den1) - logf(dd / den2));
    }
    __shared__ float red[256];
    red[threadIdx.x] = l; __syncthreads();
    for (int s = 128; s > 0; s >>= 1) { if (threadIdx.x < s) red[threadIdx.x] += red[threadIdx.x + s]; __syncthreads(); }
    if (threadIdx.x == 0) atomicAdd(&accum[0], red[0]);
}

// =====================================================================
// misc elementwise / small kernels
// =====================================================================
__global__ __launch_bounds__(128) void gather_src(
        const float* __restrict__ n1, const float* __restrict__ n2,
        const int* __restrict__ ids, float* __restrict__ s1, float* __restrict__ s2) {
    int i = blockIdx.x, t = threadIdx.x;
    int id = ids[i];
    s1[(size_t)i * D_ + t] = n1[(size_t)id * D_ + t];
    s2[(size_t)i * D_ + t] = n2[(size_t)id * D_ + t];
}

__global__ __launch_bounds__(128) void rownorm(float* __restrict__ z) {
    int r = blockIdx.x, t = threadIdx.x;
    float v = z[(size_t)r * D_ + t];
    __shared__ float red[128];
    red[t] = v * v; __syncthreads();
    for (int s = 64; s > 0; s >>= 1) { if (t < s) red[t] += red[t + s]; __syncthreads(); }
    float nrm = fmaxf(sqrtf(red[0]), 1e-12f);
    z[(size_t)r * D_ + t] = v / nrm;
}

__global__ __launch_bounds__(256) void sumsq(const float* __restrict__ x, long long n,
                                             float* __restrict__ accum) {
    long long i = (long long)blockIdx.x * 256 + threadIdx.x;
    float v = (i < n) ? x[i] : 0.0f;
    __shared__ float red[256];
    red[threadIdx.x] = v * v; __syncthreads();
    for (int s = 128; s > 0; s >>= 1) { if (threadIdx.x < s) red[threadIdx.x] += red[threadIdx.x + s]; __syncthreads(); }
    if (threadIdx.x == 0) atomicAdd(accum, red[0]);
}

__global__ __launch_bounds__(256) void relsq(const float* __restrict__ relation,
                                             const int* __restrict__ etype,
                                             float* __restrict__ accum) {
    int i = blockIdx.x, d = threadIdx.x;
    int et = etype[i];
    float v = relation[(size_t)(et % NR_) * (2 * D_) + d];
    __shared__ float red[256];
    red[d] = v * v; __syncthreads();
    for (int s = 128; s > 0; s >>= 1) { if (d < s) red[d] += red[d + s]; __syncthreads(); }
    if (d == 0) atomicAdd(accum, red[0]);
}

// msg = relu(concat(src1,src2) + sign*relation[etype%NR])
__global__ void build_msg(const float* __restrict__ s1, const float* __restrict__ s2,
                          const float* __restrict__ relation, const int* __restrict__ etype,
                          float* __restrict__ msg) {
    int gid = blockIdx.x * blockDim.x + threadIdx.x;
    if (gid >= NSRC_ * 2 * D_) return;
    int i = gid >> 8;           // 256 per row
    int d = gid & 255;
    float s = (d < D_) ? s1[(size_t)i * D_ + d] : s2[(size_t)i * D_ + (d - D_)];
    int et = etype[i];
    float sg = (et >= NR_) ? -1.0f : 1.0f;
    float m = s + sg * relation[(size_t)(et % NR_) * (2 * D_) + d];
    msg[gid] = (m > 0.0f) ? m : 0.0f;
}

// one block per (b,h): 8x8 attention over the DEG mailbox
__global__ __launch_bounds__(128) void mha_attn(
        const float* __restrict__ q, const float* __restrict__ k,
        const float* __restrict__ v, float* __restrict__ o) {
    int bh = blockIdx.x, b = bh >> 3, h = bh & 7;
    int t = threadIdx.x;
    __shared__ float qs[8][128], ks[8][128], vs[8][128], sc[8][8], inv[8];
    size_t base = (size_t)(b * DEG_) * T_ + (size_t)h * DH_;
    for (int l = 0; l < 8; ++l) {
        qs[l][t] = q[base + (size_t)l * T_ + t];
        ks[l][t] = k[base + (size_t)l * T_ + t];
        vs[l][t] = v[base + (size_t)l * T_ + t];
    }
    __syncthreads();
    if (t < 64) {
        int l = t >> 3, m = t & 7;
        float s = 0.0f;
        for (int d = 0; d < 128; ++d) s += qs[l][d] * ks[m][d];
        sc[l][m] = s * 0.08838834764831845f;   // 1/sqrt(128)
    }
    __syncthreads();
    if (t < 8) {
        float mx = sc[t][0];
        for (int m = 1; m < 8; ++m) mx = fmaxf(mx, sc[t][m]);
        float sum = 0.0f;
        for (int m = 0; m < 8; ++m) { float e = expf(sc[t][m] - mx); sc[t][m] = e; sum += e; }
        inv[t] = 1.0f / sum;
    }
    __syncthreads();
    for (int l = 0; l < 8; ++l) {
        float a = 0.0f;
        for (int m = 0; m < 8; ++m) a += sc[l][m] * vs[m][t];
        o[base + (size_t)l * T_ + t] = a * inv[l];
    }
}

// gate = softmax(o @ gate_w) per row, E=4
__global__ __launch_bounds__(128) void gate_kernel(const float* __restrict__ o,
                                                   const float* __restrict__ gw,
                                                   float* __restrict__ gate) {
    int r = blockIdx.x, t = threadIdx.x;
    float a[E_] = {0, 0, 0, 0};
    for (int u = t; u < T_; u += 128) {
        float x = o[(size_t)r * T_ + u];
        const float* w = gw + (size_t)u * E_;
        a[0] += x * w[0]; a[1] += x * w[1]; a[2] += x * w[2]; a[3] += x * w[3];
    }
    __shared__ float red[128][E_];
    for (int e = 0; e < E_; ++e) red[t][e] = a[e];
    __syncthreads();
    for (int s = 64; s > 0; s >>= 1) {
        if (t < s) for (int e = 0; e < E_; ++e) red[t][e] += red[t + s][e];
        __syncthreads();
    }
    if (t == 0) {
        float mx = fmaxf(fmaxf(red[0][0], red[0][1]), fmaxf(red[0][2], red[0][3]));
        float ex[E_], sum = 0.0f;
        for (int e = 0; e < E_; ++e) { ex[e] = expf(red[0][e] - mx); sum += ex[e]; }
        for (int e = 0; e < E_; ++e) gate[r * E_ + e] = ex[e] / sum;
    }
}

__global__ void mean_sigmoid(const float* __restrict__ moe, float* __restrict__ out) {
    int gid = blockIdx.x * blockDim.x + threadIdx.x;
    if (gid >= B_ * T_) return;
    int bb = gid / T_, tt = gid % T_;
    float s = 0.0f;
    for (int l = 0; l < DEG_; ++l) s += moe[(size_t)(bb * DEG_ + l) * T_ + tt];
    s *= (1.0f / DEG_);
    out[gid] = 1.0f / (1.0f + expf(-s));
}

__global__ void final_loss(const float* __restrict__ accum, float* __restrict__ out_loss) {
    if (blockIdx.x == 0 && threadIdx.x == 0)
        out_loss[0] = CLW_ * (accum[0] / (float)NSRC_)
                    + DECAY_ * 0.5f * accum[1] / (float)NSRC_;
}

// =====================================================================
// host launcher
// =====================================================================
static inline int cdiv(long long a, long long b) { return (int)((a + b - 1) / b); }

extern "C" void kernel_launch(void* const* d_in, const int* in_sizes, int n_in,
                              void* d_out, int out_size, void* d_ws, size_t ws_size,
                              hipStream_t stream) {
    const float* entity   = (const float*)d_in[0];
    const float* type_e   = (const float*)d_in[1];
    const float* cluster  = (const float*)d_in[2];
    const float* relation = (const float*)d_in[3];
    const float* ln_g     = (const float*)d_in[4];
    const float* ln_b     = (const float*)d_in[5];
    const float* cl_w1    = (const float*)d_in[6];
    const float* cl_b1    = (const float*)d_in[7];
    const float* cl_w2    = (const float*)d_in[8];
    const float* cl_b2    = (const float*)d_in[9];
    const float* fc_w     = (const float*)d_in[10];
    const float* fc_b     = (const float*)d_in[11];
    const float* wq       = (const float*)d_in[12];
    const float* bq       = (const float*)d_in[13];
    const float* wk       = (const float*)d_in[14];
    const float* bk_      = (const float*)d_in[15];
    const float* wv       = (const float*)d_in[16];
    const float* bv       = (const float*)d_in[17];
    const float* wo       = (const float*)d_in[18];
    const float* bo       = (const float*)d_in[19];
    const float* gate_w   = (const float*)d_in[20];
    const float* exp_w    = (const float*)d_in[21];
    const float* exp_b    = (const float*)d_in[22];
    const float* e2t_val  = (const float*)d_in[23];
    const float* t2c_val  = (const float*)d_in[24];
    const float* e2c_val  = (const float*)d_in[25];
    const int*   e2t_row  = (const int*)d_in[26];
    const int*   e2t_col  = (const int*)d_in[27];
    const int*   t2c_row  = (const int*)d_in[28];
    const int*   t2c_col  = (const int*)d_in[29];
    const int*   e2c_row  = (const int*)d_in[30];
    const int*   e2c_col  = (const int*)d_in[31];
    const int*   src_ids  = (const int*)d_in[32];
    const int*   etype    = (const int*)d_in[33];
    const int NNZ_E2T = in_sizes[23], NNZ_T2C = in_sizes[24], NNZ_E2C = in_sizes[25];

    // ---------------- workspace carve (recomputed identically each call) --
    char* w = (char*)d_ws;
    auto carve = [&](size_t bytes) { void* p = (void*)w; w += (bytes + 255) & ~(size_t)255; return p; };
    const size_t NODE_B  = (size_t)NNODES_ * D_ * sizeof(float);
    const size_t GCNMAX  = (size_t)(NE_ + NT_) * D_ * sizeof(float);
    const size_t ROW_T_B = (size_t)NSRC_ * T_ * sizeof(float);
    const size_t MHA_TOT = (size_t)NSRC_ * 2 * D_ * sizeof(float) + 7 * ROW_T_B;
    float* node1 = (float*)carve(NODE_B);
    float* node2 = (float*)carve(NODE_B);
    char*  uni   = (char*)carve((3 * GCNMAX > MHA_TOT) ? 3 * GCNMAX : MHA_TOT);
    // GCN view of union region
    float* gx   = (float*)uni;
    float* gy   = (float*)(uni + GCNMAX);
    float* gacc = (float*)(uni + 2 * GCNMAX);
    // MHA/MoE view of union region (used only after all GCN work is done)
    float* msg  = (float*)uni;
    float* pred = (float*)(uni + (size_t)NSRC_ * 2 * D_ * sizeof(float));
    float* qb   = pred + (size_t)NSRC_ * T_;
    float* kb   = qb   + (size_t)NSRC_ * T_;
    float* vb   = kb   + (size_t)NSRC_ * T_;
    float* ab   = vb   + (size_t)NSRC_ * T_;   // attention output
    float* ob   = ab   + (size_t)NSRC_ * T_;   // o = attn @ wo + bo
    float* moe  = ob   + (size_t)NSRC_ * T_;
    float* src1 = (float*)carve((size_t)NSRC_ * D_ * sizeof(float));
    float* src2 = (float*)carve((size_t)NSRC_ * D_ * sizeof(float));
    float* z1   = (float*)carve((size_t)NSRC_ * D_ * sizeof(float));
    float* z2   = (float*)carve((size_t)NSRC_ * D_ * sizeof(float));
    float* hbuf = (float*)carve((size_t)NSRC_ * D_ * sizeof(float));
    float* redb = (float*)carve(7 * (size_t)NSRC_ * sizeof(float));
    float* r11 = redb, *r12 = redb + NSRC_, *r22 = redb + 2 * NSRC_, *c12 = redb + 3 * NSRC_;
    float* d11 = redb + 4 * NSRC_, *d12 = redb + 5 * NSRC_, *d22 = redb + 6 * NSRC_;
    float* accum = (float*)carve(256);          // [0]=cl sum, [1]=sq sum
    float* gate  = (float*)carve((size_t)NSRC_ * E_ * sizeof(float));

    float* out     = (float*)d_out;             // [512,1024]
    float* out_aux = out + (size_t)B_ * T_;     // scalar

    auto gemm = [&](const float* A, int lda, const float* Bw, int ldb, const float* bias,
                    float* C, int ldc, int M, int N, int K, int act,
                    const float* gp, int expert) {
        dim3 g(N / GEMM_BN, M / GEMM_BM);
        gemm_wmma_f32<<<g, 256, 0, stream>>>(A, lda, Bw, ldb, bias, C, ldc, M, N, K, act, gp, expert);
    };

    // ---------------- 1) three LightGCN passes + LN, assembled into node1/2
    auto run_gcn = [&](const float* ea, const float* eb, int na, int nb,
                       const int* row, const int* col, const float* val, int nnz,
                       float* dstA, float* dstB) {
        long long ntot = (long long)(na + nb);
        long long nel  = ntot * D_;
        gcn_init<<<cdiv(nel, 256), 256, 0, stream>>>(ea, (long long)na * D_, eb, (long long)nb * D_, gx, gacc);
        float* xc = gx; float* yc = gy;
        for (int l = 0; l < 2; ++l) {
            hipMemsetAsync(yc, 0, nel * sizeof(float), stream);
            gcn_scatter<<<cdiv((long long)nnz * 32, 256), 256, 0, stream>>>(xc, yc, row, col, val, nnz);
            vec_add<<<cdiv(nel, 256), 256, 0, stream>>>(gacc, yc, nel);
            float* t = xc; xc = yc; yc = t;
        }
        gcn_finalize_ln<<<(int)ntot, 128, 0, stream>>>(gacc, na, ln_g, ln_b, dstA, dstB);
    };
    // node1 = [e2t_e, e2t_t, t2c_c]; node2 = [e2c_e, t2c_t, e2c_c]
    run_gcn(entity, type_e,  NE_, NT_, e2t_row, e2t_col, e2t_val, NNZ_E2T,
            node1, node1 + (size_t)NE_ * D_);
    run_gcn(type_e, cluster, NT_, NC_, t2c_row, t2c_col, t2c_val, NNZ_T2C,
            node2 + (size_t)NE_ * D_, node1 + (size_t)(NE_ + NT_) * D_);
    run_gcn(entity, cluster, NE_, NC_, e2c_row, e2c_col, e2c_val, NNZ_E2C,
            node2, node2 + (size_t)(NE_ + NT_) * D_);

    // ---------------- 2) gather + contrastive loss ----------------
    gather_src<<<NSRC_, 128, 0, stream>>>(node1, node2, src_ids, src1, src2);
    // z1 = norm(elu(src1@w1+b1)@w2+b2)
    gemm(src1, D_, cl_w1, D_, cl_b1, hbuf, D_, NSRC_, D_, D_, 1, nullptr, -1);
    gemm(hbuf, D_, cl_w2, D_, cl_b2, z1,   D_, NSRC_, D_, D_, 0, nullptr, -1);
    rownorm<<<NSRC_, 128, 0, stream>>>(z1);
    gemm(src2, D_, cl_w1, D_, cl_b1, hbuf, D_, NSRC_, D_, D_, 1, nullptr, -1);
    gemm(hbuf, D_, cl_w2, D_, cl_b2, z2,   D_, NSRC_, D_, D_, 0, nullptr, -1);
    rownorm<<<NSRC_, 128, 0, stream>>>(z2);
    hipMemsetAsync(redb, 0, 7 * (size_t)NSRC_ * sizeof(float), stream);
    hipMemsetAsync(accum, 0, 256, stream);
    { dim3 g(NSRC_ / 16, NSRC_ / 16);
      cl_sim<<<g, 32, 0, stream>>>(z1, z2, r11, r12, r22, c12, d11, d12, d22); }
    cl_finalize<<<cdiv(NSRC_, 256), 256, 0, stream>>>(r11, d11, r12, r22, d22, c12, d12, accum);
    sumsq<<<cdiv((long long)NSRC_ * D_, 256), 256, 0, stream>>>(src1, (long long)NSRC_ * D_, accum + 1);
    sumsq<<<cdiv((long long)NSRC_ * D_, 256), 256, 0, stream>>>(src2, (long long)NSRC_ * D_, accum + 1);
    relsq<<<NSRC_, 256, 0, stream>>>(relation, etype, accum + 1);

    // ---------------- 3) message + fc + MHA + MoE ----------------
    build_msg<<<cdiv((long long)NSRC_ * 2 * D_, 256), 256, 0, stream>>>(src1, src2, relation, etype, msg);
    gemm(msg, 2 * D_, fc_w, T_, fc_b, pred, T_, NSRC_, T_, 2 * D_, 0, nullptr, -1);
    gemm(pred, T_, wq, T_, bq,  qb, T_, NSRC_, T_, T_, 0, nullptr, -1);
    gemm(pred, T_, wk, T_, bk_, kb, T_, NSRC_, T_, T_, 0, nullptr, -1);
    gemm(pred, T_, wv, T_, bv,  vb, T_, NSRC_, T_, T_, 0, nullptr, -1);
    mha_attn<<<B_ * H_, 128, 0, stream>>>(qb, kb, vb, ab);
    gemm(ab, T_, wo, T_, bo, ob, T_, NSRC_, T_, T_, 0, nullptr, -1);
    gate_kernel<<<NSRC_, 128, 0, stream>>>(ob, gate_w, gate);
    hipMemsetAsync(moe, 0, ROW_T_B, stream);
    for (int e = 0; e < E_; ++e)
        gemm(ob, T_, exp_w + (size_t)e * T_ * T_, T_, exp_b + (size_t)e * T_,
             moe, T_, NSRC_, T_, T_, 0, gate, e);
    mean_sigmoid<<<cdiv((long long)B_ * T_, 256), 256, 0, stream>>>(moe, out);
    final_loss<<<1, 1, 0, stream>>>(accum, out_aux);
}